// HyperGraphConvolution_22866405884230
// MI455X (gfx1250) — compile-verified
//
#include <hip/hip_runtime.h>

typedef __attribute__((ext_vector_type(2))) float v2f;
typedef __attribute__((ext_vector_type(8))) float v8f;

// ---------------------------------------------------------------------------
// Zero-fill (float4 vectorized, grid-stride)
// ---------------------------------------------------------------------------
__global__ void fill_zero_f4(float4* __restrict__ p, long n4) {
    long i = (long)blockIdx.x * blockDim.x + threadIdx.x;
    long stride = (long)gridDim.x * blockDim.x;
    float4 z = make_float4(0.f, 0.f, 0.f, 0.f);
    for (; i < n4; i += stride) p[i] = z;
}

// ---------------------------------------------------------------------------
// Scatter SpMM: dst[rows[e], col_off + d] += vals[e] * src[cols[e], d]
// src row stride is 64; one thread per (edge, dim) pair (d = 0..63).
// ---------------------------------------------------------------------------
__global__ void scatter_spmm(const int* __restrict__ rows,
                             const int* __restrict__ cols,
                             const float* __restrict__ vals,
                             const float* __restrict__ src,
                             float* __restrict__ dst,
                             int dst_ld, int col_off, long nE) {
    long gid = (long)blockIdx.x * blockDim.x + threadIdx.x;
    long e = gid >> 6;
    int  d = (int)(gid & 63);
    if (e >= nE) return;
    int   r = rows[e];
    int   c = cols[e];
    float v = vals[e];
    atomicAdd(&dst[(long)r * dst_ld + col_off + d], v * src[(long)c * 64 + d]);
}

// ---------------------------------------------------------------------------
// msg[NGx64] = gmsg[NGx128] @ Wl[128x64] + bl ; out_g += msg   (fused)
// Block = 128 threads (4 waves). Block owns 16 rows (one M tile).
// Wave w owns N tile [16w, 16w+16). K = 128 -> 32 x V_WMMA_F32_16X16X4_F32.
//
// Fragment layouts per CDNA5 ISA 7.12.2 (wave32):
//   A (16x4 f32):  lane&15 = M, (lane>>4)*2 = K base; v.x = K, v.y = K+1
//   B (4x16 f32):  lane&15 = N, (lane>>4)*2 = K base; v.x = K, v.y = K+1
//   C/D (16x16 f32): VGPR r -> M = r (lanes 0-15) / r+8 (lanes 16-31), N = lane&15
// ---------------------------------------------------------------------------
__global__ void __launch_bounds__(128)
gemm_msg_wmma(const float* __restrict__ gmsg,
              const float* __restrict__ Wl,
              const float* __restrict__ bl,
              float* __restrict__ msg,
              float* __restrict__ out_g,
              int ngroups) {
    __shared__ float Atile[16 * 128];   // 8 KB A staging

    const int tid   = threadIdx.x;
    const int mbase = blockIdx.x * 16;

    // Stage the 16x128 A tile into LDS (512 float4 / 128 threads = 4 each).
    for (int j = 0; j < 4; ++j) {
        int f4 = tid + j * 128;            // 0..511
        int r  = f4 >> 5;                  // row    0..15
        int k  = (f4 & 31) << 2;           // column 0..124, step 4
        int grow = mbase + r;
        if (grow >= ngroups) grow = ngroups - 1;   // clamped (values unused on tail)
        const float4 a = *reinterpret_cast<const float4*>(&gmsg[(long)grow * 128 + k]);
        *reinterpret_cast<float4*>(&Atile[r * 128 + k]) = a;
    }
    __syncthreads();

    const int lane = tid & 31;
    const int wave = tid >> 5;              // N tile index, 0..3
    const int mrow = lane & 15;
    const int kb   = (lane >> 4) << 1;      // 0 or 2
    const int ncol = (wave << 4) + (lane & 15);

    v8f c = {};
    #pragma unroll
    for (int k0 = 0; k0 < 128; k0 += 4) {
        v2f a, b;
        a.x = Atile[mrow * 128 + k0 + kb];
        a.y = Atile[mrow * 128 + k0 + kb + 1];
        b.x = Wl[(k0 + kb) * 64 + ncol];
        b.y = Wl[(k0 + kb + 1) * 64 + ncol];
        c = __builtin_amdgcn_wmma_f32_16x16x4_f32(
                /*neg_a=*/false, a, /*neg_b=*/false, b,
                /*c_mod=*/(short)0, c, /*reuse_a=*/false, /*reuse_b=*/false);
    }

    const float bias = bl[ncol];
    const int   moff = (lane >> 4) << 3;    // 0 or 8
    #pragma unroll
    for (int r = 0; r < 8; ++r) {
        int row = mbase + moff + r;
        if (row < ngroups) {
            float vv  = c[r] + bias;
            long  idx = (long)row * 64 + ncol;
            msg[idx]   = vv;
            out_g[idx] += vv;
        }
    }
}

// ---------------------------------------------------------------------------
// dst += src  (float4 vectorized, grid-stride)
// ---------------------------------------------------------------------------
__global__ void accum_f4(float4* __restrict__ dst, const float4* __restrict__ src, long n4) {
    long i = (long)blockIdx.x * blockDim.x + threadIdx.x;
    long stride = (long)gridDim.x * blockDim.x;
    for (; i < n4; i += stride) {
        float4 d = dst[i];
        float4 s = src[i];
        d.x += s.x; d.y += s.y; d.z += s.z; d.w += s.w;
        dst[i] = d;
    }
}

// ---------------------------------------------------------------------------
// Host-side orchestration
// ---------------------------------------------------------------------------
extern "C" void kernel_launch(void* const* d_in, const int* in_sizes, int n_in,
                              void* d_out, int out_size, void* d_ws, size_t ws_size,
                              hipStream_t stream) {
    (void)n_in; (void)out_size; (void)ws_size;

    const float* user_emb  = (const float*)d_in[0];
    const float* item_emb  = (const float*)d_in[1];
    const float* group_emb = (const float*)d_in[2];
    const float* u_vals    = (const float*)d_in[3];
    const float* i_vals    = (const float*)d_in[4];
    const float* f_vals    = (const float*)d_in[5];
    const float* W         = (const float*)d_in[6];
    const float* b         = (const float*)d_in[7];
    const int*   u_rows    = (const int*)d_in[8];
    const int*   u_cols    = (const int*)d_in[9];
    const int*   i_rows    = (const int*)d_in[10];
    const int*   i_cols    = (const int*)d_in[11];
    const int*   f_rows    = (const int*)d_in[12];
    const int*   f_cols    = (const int*)d_in[13];

    const int  D  = 64;
    const long NU = in_sizes[0] / D;
    const long NI = in_sizes[1] / D;
    const long NG = in_sizes[2] / D;
    const long EU = in_sizes[3];
    const long EI = in_sizes[4];
    const long EF = in_sizes[5];
    const int  L  = in_sizes[6] / (2 * D * D);
    const long NN = NU + NI;

    float* out_ui = (float*)d_out;            // [NN, 64]
    float* out_g  = out_ui + NN * D;          // [NG, 64]

    // Workspace layout
    float* gmsg = (float*)d_ws;               // [NG, 128]
    float* msg  = gmsg + NG * 128;            // [NG, 64]
    float* emb  = msg  + NG * D;              // [NN, 64]

    // Initialize accumulators: final_ui = concat(user, item); final_g = group.
    hipMemcpyAsync(out_ui,          user_emb,  (size_t)NU * D * sizeof(float),
                   hipMemcpyDeviceToDevice, stream);
    hipMemcpyAsync(out_ui + NU * D, item_emb,  (size_t)NI * D * sizeof(float),
                   hipMemcpyDeviceToDevice, stream);
    hipMemcpyAsync(out_g,           group_emb, (size_t)NG * D * sizeof(float),
                   hipMemcpyDeviceToDevice, stream);

    const float* uemb = user_emb;
    const float* iemb = item_emb;

    for (int l = 0; l < L; ++l) {
        // gmsg = 0
        fill_zero_f4<<<1024, 256, 0, stream>>>((float4*)gmsg, NG * 128 / 4);

        // gmsg[:, 0:64]   += scatter(user edges)
        long tu = EU * 64;
        scatter_spmm<<<(int)((tu + 255) / 256), 256, 0, stream>>>(
            u_rows, u_cols, u_vals, uemb, gmsg, 128, 0, EU);

        // gmsg[:, 64:128] += scatter(item edges)
        long ti = EI * 64;
        scatter_spmm<<<(int)((ti + 255) / 256), 256, 0, stream>>>(
            i_rows, i_cols, i_vals, iemb, gmsg, 128, 64, EI);

        // msg = gmsg @ W[l] + b[l];  out_g += msg   (fp32 WMMA)
        int mtiles = (int)((NG + 15) / 16);
        gemm_msg_wmma<<<mtiles, 128, 0, stream>>>(
            gmsg, W + (long)l * 128 * 64, b + (long)l * 64, msg, out_g, (int)NG);

        // emb = 0; emb = scatter(fan-out edges, msg)
        fill_zero_f4<<<2048, 256, 0, stream>>>((float4*)emb, NN * 64 / 4);
        long tf = EF * 64;
        scatter_spmm<<<(int)((tf + 255) / 256), 256, 0, stream>>>(
            f_rows, f_cols, f_vals, msg, emb, 64, 0, EF);

        // final_ui += emb
        accum_f4<<<2048, 256, 0, stream>>>((float4*)out_ui, (const float4*)emb, NN * 64 / 4);

        // next layer reads this layer's node embeddings
        uemb = emb;
        iemb = emb + NU * D;
    }
}